// QuantumLayer_88613765251801
// MI455X (gfx1250) — compile-verified
//
#include <hip/hip_runtime.h>

// 8-qubit batched statevector simulator for MI455X (gfx1250, wave32).
// One wave per batch element. State = 16x16 complex f32 tile in LDS
// (rows = qubits 0-3, cols = qubits 4-7). Single-qubit mixing gates are
// 16x16x16 f32 GEMMs on the matrix pipe (v_wmma_f32_16x16x4_f32, 4 K-chunks).
// Permutation/phase gates (PX/PY/PZ/CNOT/CZ/CSWAP/TOF) are LDS gathers with
// sign/i twiddles. All design-dependent control flow is scalar (SMEM loads +
// s_cbranch), so EXEC is all-ones at every WMMA as the ISA requires.

typedef float v8f __attribute__((ext_vector_type(8)));
typedef float v2f __attribute__((ext_vector_type(2)));

#define NQ     8
#define DEPTH  6
#define PITCH  17      // 16 + 1 padding -> bank-conflict-free columns
#define WPB    8       // waves (batch elements) per 256-thread block

// G16[rp][rc] for 2x2 gate g acting on bit q of a 4-bit subspace index.
__device__ __forceinline__ float gentry(int rp, int rc, int q,
                                        float g00, float g01,
                                        float g10, float g11) {
  const int mask = 1 << q;
  if ((rp ^ rc) & ~mask) return 0.0f;
  const int bp = (rp >> q) & 1, bc = (rc >> q) & 1;
  return bp ? (bc ? g11 : g10) : (bc ? g01 : g00);
}

// Apply complex 2x2 gate on qubit `q`.
// q in 0..3: S' = G16 @ S ; q in 4..7: S' = (G16 @ S^T)^T.
// Both realized via scalar strides (sk, sn) into the LDS tile -> no divergence.
template <bool CPLX>
__device__ __forceinline__ void apply_1q(float* Re, float* Im,
                                         int lane, int q,
                                         float r00, float r01, float r10, float r11,
                                         float i00, float i01, float i10, float i11) {
  const int  n       = lane & 15;   // B/C/D column index, also A row (M)
  const int  h       = lane >> 4;   // half-wave selector
  const bool rowside = (q < 4);
  const int  qe      = rowside ? q : (q - 4);
  const int  sk      = rowside ? PITCH : 1;  // stride along the acted dimension
  const int  sn      = rowside ? 1 : PITCH;  // stride along the spectator dim

  v2f br[4], bi[4], ar[4], ai[4];
#pragma unroll
  for (int k = 0; k < 4; ++k) {
    const int k0 = 4 * k + 2 * h;
    const int k1 = k0 + 1;
    // B operand chunk: rows k0,k1 of M (M = S or S^T via strides)
    br[k][0] = Re[k0 * sk + n * sn]; br[k][1] = Re[k1 * sk + n * sn];
    bi[k][0] = Im[k0 * sk + n * sn]; bi[k][1] = Im[k1 * sk + n * sn];
    // A operand chunk: G16[n][k0], G16[n][k1]
    ar[k][0] = gentry(n, k0, qe, r00, r01, r10, r11);
    ar[k][1] = gentry(n, k1, qe, r00, r01, r10, r11);
    if (CPLX) {
      ai[k][0] = gentry(n, k0, qe, i00, i01, i10, i11);
      ai[k][1] = gentry(n, k1, qe, i00, i01, i10, i11);
    }
  }

  v8f accR = {0.f, 0.f, 0.f, 0.f, 0.f, 0.f, 0.f, 0.f};
  v8f accI = {0.f, 0.f, 0.f, 0.f, 0.f, 0.f, 0.f, 0.f};

  // Re(out) = Gr*Mr ; Im(out) = Gr*Mi
#pragma unroll
  for (int k = 0; k < 4; ++k) {
    accR = __builtin_amdgcn_wmma_f32_16x16x4_f32(false, ar[k], false, br[k],
                                                 (short)0, accR, false, false);
    accI = __builtin_amdgcn_wmma_f32_16x16x4_f32(false, ar[k], false, bi[k],
                                                 (short)0, accI, false, false);
  }
  // Re(out) -= Gi*Mi ; Im(out) += Gi*Mr
  if (CPLX) {
#pragma unroll
    for (int k = 0; k < 4; ++k) {
      v2f nai = -ai[k];
      accR = __builtin_amdgcn_wmma_f32_16x16x4_f32(false, nai, false, bi[k],
                                                   (short)0, accR, false, false);
      accI = __builtin_amdgcn_wmma_f32_16x16x4_f32(false, ai[k], false, br[k],
                                                   (short)0, accI, false, false);
    }
  }

  // Scatter D (C-layout: acted index = v + 8h, spectator = n) back.
#pragma unroll
  for (int v = 0; v < 8; ++v) {
    const int m = v + 8 * h;
    Re[m * sk + n * sn] = accR[v];
    Im[m * sk + n * sn] = accI[v];
  }
}

// Permutation / phase gates. Self-inverse permutations: gather from perm(i),
// scale by {+-1, +-i}.
__device__ __forceinline__ void apply_perm(float* Re, float* Im,
                                           int lane, int g2, int node) {
  const int a = node, b = (node + 1) & 7, c = (node + 2) & 7;
  float tR[8], tI[8];
#pragma unroll
  for (int j = 0; j < 8; ++j) {
    const int i = lane + 32 * j;
    int src = i;
    float mR = 1.0f, mI = 0.0f;
    switch (g2) {
      case 2:  src = i ^ (1 << a); break;                               // PX
      case 3:  src = i ^ (1 << a);                                      // PY
               if ((i >> a) & 1) { mR = 0.f; mI =  1.f; }
               else              { mR = 0.f; mI = -1.f; }
               break;
      case 4:  if ((i >> a) & 1) mR = -1.f; break;                      // PZ
      case 5:  if ((i >> a) & 1) src = i ^ (1 << b); break;             // CNOT
      case 8:  if (((i >> a) & 1) && ((i >> b) & 1)) mR = -1.f; break;  // CZ
      case 6:  if ((i >> a) & 1) {                                      // CSWAP
                 const int bb = (i >> b) & 1, cc = (i >> c) & 1;
                 if (bb != cc) src = i ^ (1 << b) ^ (1 << c);
               }
               break;
      case 7:  if (((i >> a) & 1) && ((i >> b) & 1)) src = i ^ (1 << c); // TOF
               break;
      default: break;
    }
    const float re = Re[(src & 15) * PITCH + (src >> 4)];
    const float im = Im[(src & 15) * PITCH + (src >> 4)];
    tR[j] = mR * re - mI * im;
    tI[j] = mR * im + mI * re;
  }
  asm volatile("s_wait_dscnt 0" ::: "memory");  // all gathers before scatter
#pragma unroll
  for (int j = 0; j < 8; ++j) {
    const int i = lane + 32 * j;
    Re[(i & 15) * PITCH + (i >> 4)] = tR[j];
    Im[(i & 15) * PITCH + (i >> 4)] = tI[j];
  }
}

__global__ __launch_bounds__(256) void qsim_kernel(
    const float* __restrict__ inF,  // [B, 8]
    const float* __restrict__ qp,   // [48]
    const int*   __restrict__ dry,  // [6,4]
    const int*   __restrict__ drot, // [6,4]
    const int*   __restrict__ dg2,  // [6,4]
    float* __restrict__ out,        // [B, 8]
    int batch) {
  __shared__ float SreB[WPB][16][PITCH];
  __shared__ float SimB[WPB][16][PITCH];

  const int lane = threadIdx.x & 31;
  const int wid  = threadIdx.x >> 5;
  // Wave-uniform batch index; readfirstlane makes the bounds check a scalar
  // branch so EXEC is never masked around the WMMA regions.
  const int bidx = __builtin_amdgcn_readfirstlane(blockIdx.x * WPB + wid);
  if (bidx >= batch) return;

  float* Re = &SreB[wid][0][0];
  float* Im = &SimB[wid][0][0];

  // H^x8 |0..0> == uniform real amplitude 1/16: fold initial Hadamards in.
#pragma unroll
  for (int j = 0; j < 8; ++j) {
    const int i = lane + 32 * j;
    Re[(i & 15) * PITCH + (i >> 4)] = 0.0625f;
    Im[(i & 15) * PITCH + (i >> 4)] = 0.0f;
  }

#pragma unroll 1
  for (int layer = 0; layer < DEPTH; ++layer) {
#pragma unroll 1
    for (int node = 0; node < NQ; ++node) {
      const int di = layer * 4 + (node & 3);  // (layer % 6, node % 4)

      if (dry[di]) {  // batched RY(theta = input_features[b, node]), real
        const float th = inF[bidx * NQ + node] * 0.5f;  // scalar load (uniform)
        const float cc = __cosf(th), ss = __sinf(th);
        apply_1q<false>(Re, Im, lane, node, cc, -ss, ss, cc, 0, 0, 0, 0);
      }

      const int rot = drot[di];
      if (rot < 3) {
        const float th = qp[layer * NQ + node] * 0.5f;
        const float cc = __cosf(th), ss = __sinf(th);
        if (rot == 0)       // RX = [[c,-is],[-is,c]]
          apply_1q<true>(Re, Im, lane, node, cc, 0, 0, cc, 0, -ss, -ss, 0);
        else if (rot == 1)  // RY = [[c,-s],[s,c]]
          apply_1q<false>(Re, Im, lane, node, cc, -ss, ss, cc, 0, 0, 0, 0);
        else                // RZ = diag(e^{-it/2}, e^{+it/2})
          apply_1q<true>(Re, Im, lane, node, cc, 0, 0, cc, -ss, 0, 0, ss);
      }

      const int g2 = dg2[di];
      if (g2 == 1) {  // H
        const float r2 = 0.70710678118654752f;
        apply_1q<false>(Re, Im, lane, node, r2, r2, r2, -r2, 0, 0, 0, 0);
      } else if (g2 >= 2 && g2 <= 8) {
        apply_perm(Re, Im, lane, g2, node);
      }
    }
  }

  // <Z_w> = sum_i (1 - 2*bit_w(i)) * |amp_i|^2
  float acc[8] = {0, 0, 0, 0, 0, 0, 0, 0};
#pragma unroll
  for (int j = 0; j < 8; ++j) {
    const int   i  = lane + 32 * j;
    const float re = Re[(i & 15) * PITCH + (i >> 4)];
    const float im = Im[(i & 15) * PITCH + (i >> 4)];
    const float p  = re * re + im * im;
#pragma unroll
    for (int q = 0; q < 8; ++q) acc[q] += ((i >> q) & 1) ? -p : p;
  }
#pragma unroll
  for (int q = 0; q < 8; ++q) {
#pragma unroll
    for (int m = 16; m >= 1; m >>= 1) acc[q] += __shfl_xor(acc[q], m, 32);
  }
  if (lane < NQ) out[bidx * NQ + lane] = acc[lane];
}

extern "C" void kernel_launch(void* const* d_in, const int* in_sizes, int n_in,
                              void* d_out, int out_size, void* d_ws, size_t ws_size,
                              hipStream_t stream) {
  const float* inF  = (const float*)d_in[0];
  const float* qp   = (const float*)d_in[1];
  const int*   dry  = (const int*)d_in[2];
  const int*   drot = (const int*)d_in[3];
  const int*   dg2  = (const int*)d_in[4];
  float*       out  = (float*)d_out;

  const int batch  = in_sizes[0] / NQ;          // 16384
  const int blocks = (batch + WPB - 1) / WPB;   // 2048 blocks of 8 waves
  qsim_kernel<<<blocks, 256, 0, stream>>>(inF, qp, dry, drot, dg2, out, batch);
}